// ChamferLoss_80092550135919
// MI455X (gfx1250) — compile-verified
//
#include <hip/hip_runtime.h>

typedef __attribute__((ext_vector_type(2))) float v2f;
typedef __attribute__((ext_vector_type(8))) float v8f;

#define NPTS 4096
#define BATCH 16
#define WAVES_PER_BLOCK 8
#define ROWS_PER_BLOCK (WAVES_PER_BLOCK * 16)           // 128
#define BLOCKS_X (NPTS / ROWS_PER_BLOCK)                // 32
#define PARTIALS_PER_DIR (BLOCKS_X * BATCH * WAVES_PER_BLOCK)  // 4096
#define TOTAL_PARTIALS (2 * PARTIALS_PER_DIR)           // 8192

// One direction of the Chamfer distance: for each row of `src`, min over all
// rows of `dst` of squared L2 distance, then sqrt, then per-wave partial sums.
// d2(n,m) = |p_n|^2 + |t_m|^2 - 2 * <p_n, t_m>, with the inner product tile
// computed by V_WMMA_F32_16X16X4_F32 (K padded 3 -> 4 with zeros).
__global__ __launch_bounds__(256)
void chamfer_dir_kernel(const float* __restrict__ src,   // (B, N, 3)
                        const float* __restrict__ dst,   // (B, M, 3)
                        float* __restrict__ partial,
                        int partial_base) {
  __shared__ float sT[NPTS * 3];                         // 48 KB of one batch's dst pts

  const int b   = blockIdx.y;
  const int tid = threadIdx.x;

  // Stage all dst points for this batch into LDS (reused by all 8 waves x 256 tiles).
  const float* dp = dst + (size_t)b * NPTS * 3;
  for (int i = tid; i < NPTS * 3; i += 256) sT[i] = dp[i];
  __syncthreads();

  const int wave = tid >> 5;
  const int lane = tid & 31;
  const int lp   = lane & 15;
  const int half = lane >> 4;

  const int n0 = (blockIdx.x * WAVES_PER_BLOCK + wave) * 16;
  const float* sp = src + ((size_t)b * NPTS + n0) * 3;

  // A fragment (16x4 f32): lanes 0-15 hold {x,y} (K=0,1) of row n0+lp,
  // lanes 16-31 hold {z,0} (K=2,3) of the same row.
  const float px = sp[3 * lp + 0];
  const float py = sp[3 * lp + 1];
  const float pz = sp[3 * lp + 2];
  v2f afrag;
  afrag.x = half ? pz : px;
  afrag.y = half ? 0.0f : py;

  // |p|^2 for row n0+lp (lanes 0-15 meaningful; lanes 16-31 duplicate).
  const float rsq_mine = px * px + py * py + pz * pz;

  // Epilogue row mapping: lane handles rows n0 + half*8 + r, r = 0..7.
  float rs[8];
#pragma unroll
  for (int r = 0; r < 8; ++r)
    rs[r] = __shfl(rsq_mine, half * 8 + r, 32);

  float minv[8];
#pragma unroll
  for (int r = 0; r < 8; ++r) minv[r] = 3.4e38f;

  for (int m0 = 0; m0 < NPTS; m0 += 16) {
    // B fragment (4x16 f32): lane's column is m0+lp; lanes 0-15 hold {x,y},
    // lanes 16-31 hold {z,0} (K split across lane halves, mirroring A).
    const float* tp = &sT[3 * (m0 + lp)];
    const float tx = tp[0];
    const float ty = tp[1];
    const float tz = tp[2];
    v2f bfrag;
    bfrag.x = half ? tz : tx;
    bfrag.y = half ? 0.0f : ty;
    const float csq = tx * tx + ty * ty + tz * tz;       // |t_{m0+lp}|^2

    v8f c = {};
    c = __builtin_amdgcn_wmma_f32_16x16x4_f32(
        /*neg_a=*/false, afrag, /*neg_b=*/false, bfrag,
        /*c_mod=*/(short)0, c, /*reuse_a=*/false, /*reuse_b=*/false);

#pragma unroll
    for (int r = 0; r < 8; ++r) {
      const float d2 = fmaf(-2.0f, c[r], rs[r] + csq);
      minv[r] = fminf(minv[r], d2);
    }
  }

  // Min-reduce each row across its 16-lane half, then sqrt of clamped min.
  float sum = 0.0f;
#pragma unroll
  for (int r = 0; r < 8; ++r) {
    float m = minv[r];
    m = fminf(m, __shfl_xor(m, 1, 32));
    m = fminf(m, __shfl_xor(m, 2, 32));
    m = fminf(m, __shfl_xor(m, 4, 32));
    m = fminf(m, __shfl_xor(m, 8, 32));
    sum += sqrtf(fmaxf(m, 0.0f));                        // rows half*8 + r
  }
  // Combine the two half-wave sums (rows 0-7 + rows 8-15).
  sum += __shfl_xor(sum, 16, 32);

  if (lane == 0) {
    const int slot =
        partial_base + (b * gridDim.x + blockIdx.x) * WAVES_PER_BLOCK + wave;
    partial[slot] = sum;
  }
}

// Deterministic fixed-order tree reduction of all per-wave partials.
__global__ __launch_bounds__(256)
void chamfer_reduce_kernel(const float* __restrict__ partial,
                           float* __restrict__ out) {
  __shared__ float s[256];
  float acc = 0.0f;
  for (int i = threadIdx.x; i < TOTAL_PARTIALS; i += 256) acc += partial[i];
  s[threadIdx.x] = acc;
  __syncthreads();
  for (int off = 128; off > 0; off >>= 1) {
    if (threadIdx.x < off) s[threadIdx.x] += s[threadIdx.x + off];
    __syncthreads();
  }
  if (threadIdx.x == 0) {
    // mean over batches of (mean_n dist1 + mean_m dist2) / 2, N == M:
    out[0] = s[0] * (1.0f / (2.0f * (float)BATCH * (float)NPTS));
  }
}

extern "C" void kernel_launch(void* const* d_in, const int* in_sizes, int n_in,
                              void* d_out, int out_size, void* d_ws,
                              size_t ws_size, hipStream_t stream) {
  const float* pred   = (const float*)d_in[0];
  const float* target = (const float*)d_in[1];
  float* partial = (float*)d_ws;                         // 8192 floats = 32 KB

  dim3 grid(BLOCKS_X, BATCH);
  // dist1: pred -> target
  chamfer_dir_kernel<<<grid, 256, 0, stream>>>(pred, target, partial, 0);
  // dist2: target -> pred
  chamfer_dir_kernel<<<grid, 256, 0, stream>>>(target, pred, partial,
                                               PARTIALS_PER_DIR);
  chamfer_reduce_kernel<<<1, 256, 0, stream>>>(partial, (float*)d_out);
}